// Model_65678639890853
// MI455X (gfx1250) — compile-verified
//
#include <hip/hip_runtime.h>
#include <cmath>

// ---------------------------------------------------------------------------
// Types for WMMA
// ---------------------------------------------------------------------------
typedef float v2f __attribute__((ext_vector_type(2)));
typedef float v8f __attribute__((ext_vector_type(8)));

#define SEQ 5
#define DIM 6
#define NLAYERS 5
#define LN_EPS 1e-12f

// ---------------------------------------------------------------------------
// Helpers for the tiny (exact, f32) transformer
// ---------------------------------------------------------------------------
__device__ inline void ln6(float* x, const float* g, const float* b) {
    float mu = 0.0f;
    for (int d = 0; d < DIM; ++d) mu += x[d];
    mu *= (1.0f / DIM);
    float var = 0.0f;
    for (int d = 0; d < DIM; ++d) { float t = x[d] - mu; var += t * t; }
    var *= (1.0f / DIM);
    float inv = rsqrtf(var + LN_EPS);
    for (int d = 0; d < DIM; ++d) x[d] = (x[d] - mu) * inv * g[d] + b[d];
}

__device__ inline float gelu_exact(float x) {
    return 0.5f * x * (1.0f + erff(x * 0.70710678118654752f));
}

// ---------------------------------------------------------------------------
// Kernel 1: 5-layer BERT on [1,5,6]; outputs v6[6] = 2*sigmoid(mean(h)) - 1
// Single lane — total work is a few kFLOPs, runs once.
// ---------------------------------------------------------------------------
__global__ void tiny_transformer_kernel(
    const float* __restrict__ vel,
    const float* __restrict__ ln0_g, const float* __restrict__ ln0_b,
    const float* __restrict__ Wq, const float* __restrict__ bq,
    const float* __restrict__ Wk, const float* __restrict__ bk,
    const float* __restrict__ Wv, const float* __restrict__ bv,
    const float* __restrict__ Wo, const float* __restrict__ bo,
    const float* __restrict__ ln1_g, const float* __restrict__ ln1_b,
    const float* __restrict__ Wi, const float* __restrict__ bi,
    const float* __restrict__ Wp, const float* __restrict__ bp,
    const float* __restrict__ ln2_g, const float* __restrict__ ln2_b,
    float* __restrict__ v6out)
{
    if (threadIdx.x != 0 || blockIdx.x != 0) return;

    float h[SEQ][DIM];

    // Embeddings: input + sinusoidal position encoding, then LayerNorm.
    for (int s = 0; s < SEQ; ++s) {
        for (int d = 0; d < DIM; ++d) {
            float expo  = (float)(2 * (d / 2)) / (float)DIM;
            float denom = powf(10000.0f, expo);
            float ang   = (float)s / denom;
            float pe    = (d & 1) ? cosf(ang) : sinf(ang);
            h[s][d] = vel[s * DIM + d] + pe;
        }
        ln6(h[s], ln0_g, ln0_b);
    }

    const float scale = 0.57735026918962576f; // 1/sqrt(DH), DH=3

    for (int l = 0; l < NLAYERS; ++l) {
        const float* wq = Wq + l * 36;  const float* bql = bq + l * 6;
        const float* wk = Wk + l * 36;  const float* bkl = bk + l * 6;
        const float* wv = Wv + l * 36;  const float* bvl = bv + l * 6;
        const float* wo = Wo + l * 36;  const float* bol = bo + l * 6;
        const float* wi = Wi + l * 36;  const float* bil = bi + l * 6;
        const float* wp = Wp + l * 36;  const float* bpl = bp + l * 6;
        const float* g1 = ln1_g + l * 6; const float* b1 = ln1_b + l * 6;
        const float* g2 = ln2_g + l * 6; const float* b2 = ln2_b + l * 6;

        float q[SEQ][DIM], k[SEQ][DIM], v[SEQ][DIM];
        for (int s = 0; s < SEQ; ++s) {
            for (int j = 0; j < DIM; ++j) {
                float aq = bql[j], ak = bkl[j], av = bvl[j];
                for (int i = 0; i < DIM; ++i) {
                    aq += h[s][i] * wq[i * DIM + j];
                    ak += h[s][i] * wk[i * DIM + j];
                    av += h[s][i] * wv[i * DIM + j];
                }
                q[s][j] = aq; k[s][j] = ak; v[s][j] = av;
            }
        }

        // Two heads of dim 3; softmax over keys.
        float ctx[SEQ][DIM];
        for (int hh = 0; hh < 2; ++hh) {
            const int off = hh * 3;
            for (int qi = 0; qi < SEQ; ++qi) {
                float sc[SEQ];
                float mx = -3.0e38f;
                for (int ki = 0; ki < SEQ; ++ki) {
                    float d0 = q[qi][off + 0] * k[ki][off + 0]
                             + q[qi][off + 1] * k[ki][off + 1]
                             + q[qi][off + 2] * k[ki][off + 2];
                    sc[ki] = d0 * scale;
                    mx = fmaxf(mx, sc[ki]);
                }
                float den = 0.0f;
                for (int ki = 0; ki < SEQ; ++ki) { sc[ki] = expf(sc[ki] - mx); den += sc[ki]; }
                float inv = 1.0f / den;
                for (int j = 0; j < 3; ++j) {
                    float acc = 0.0f;
                    for (int ki = 0; ki < SEQ; ++ki) acc += sc[ki] * v[ki][off + j];
                    ctx[qi][off + j] = acc * inv;
                }
            }
        }

        // Output projection + residual + LN1
        float tmp[SEQ][DIM];
        for (int s = 0; s < SEQ; ++s) {
            for (int j = 0; j < DIM; ++j) {
                float a = h[s][j] + bol[j];
                for (int i = 0; i < DIM; ++i) a += ctx[s][i] * wo[i * DIM + j];
                tmp[s][j] = a;
            }
        }
        for (int s = 0; s < SEQ; ++s) {
            ln6(tmp[s], g1, b1);
            for (int j = 0; j < DIM; ++j) h[s][j] = tmp[s][j];
        }

        // FFN (exact GELU) + residual + LN2
        float ff[SEQ][DIM];
        for (int s = 0; s < SEQ; ++s) {
            for (int j = 0; j < DIM; ++j) {
                float a = bil[j];
                for (int i = 0; i < DIM; ++i) a += h[s][i] * wi[i * DIM + j];
                ff[s][j] = gelu_exact(a);
            }
        }
        for (int s = 0; s < SEQ; ++s) {
            for (int j = 0; j < DIM; ++j) {
                float a = h[s][j] + bpl[j];
                for (int i = 0; i < DIM; ++i) a += ff[s][i] * wp[i * DIM + j];
                tmp[s][j] = a;
            }
        }
        for (int s = 0; s < SEQ; ++s) {
            ln6(tmp[s], g2, b2);
            for (int j = 0; j < DIM; ++j) h[s][j] = tmp[s][j];
        }
    }

    // pooled = sigmoid(mean over sequence); v6 = 2*pooled - 1
    for (int d = 0; d < DIM; ++d) {
        float m = 0.0f;
        for (int s = 0; s < SEQ; ++s) m += h[s][d];
        m *= (1.0f / SEQ);
        float pooled = 1.0f / (1.0f + expf(-m));
        v6out[d] = 2.0f * pooled - 1.0f;
    }
}

// ---------------------------------------------------------------------------
// Kernel 2: f_hat[plane, p] = dot(Ls[p, 0:6], v6) via V_WMMA_F32_16X16X4_F32.
//
// gridDim.y selects the plane (Lsx / Lsy) so the inner loop is a pure linear
// stream. Each wave processes TWO 16-pixel tiles per iteration (32 pixels):
//   A (16x4 f32): lanes 0-15 hold {K0,K1} of pixel M=lane, lanes 16-31 hold
//     {K2,K3} of pixel M=lane-16 -> one b64 load per lane per tile; a second
//     half-EXEC b64 load supplies {K4,K5} for the K=4..7 (zero-padded) step.
//   B (4x16 f32): v6[k] replicated across columns (rows striped across lane
//     halves: VGPR0=K0|K2, VGPR1=K1|K3).
//   Two chained WMMAs per tile accumulate the 6-term dot.
// D layout (vgpr r = row r / r+8 per lane half, all columns equal) lets the
// 32 lanes cover the 32 consecutive outputs with one full-EXEC b32 store:
//   perm(l) = l<8 ? l : l<16 ? l+8 : l<24 ? l-8 : l
//   lanes 0-7/16-23 read tile0's fragment, lanes 8-15/24-31 read tile1's.
// ---------------------------------------------------------------------------
__global__ __launch_bounds__(256) void field_project_kernel(
    const float* __restrict__ Lsx,
    const float* __restrict__ Lsy,
    const float* __restrict__ v6ws,
    float* __restrict__ out,
    int npix)
{
    const int  lane          = (int)(threadIdx.x & 31u);
    const int  wavesPerBlock = (int)(blockDim.x >> 5);
    const long nWaves        = (long)gridDim.x * wavesPerBlock;
    const long wid0          = (long)blockIdx.x * wavesPerBlock + (int)(threadIdx.x >> 5);
    const long stridePix     = nWaves * 32;

    const int plane = (int)blockIdx.y;
    const float* __restrict__ src = plane ? Lsy : Lsx;
    float*       __restrict__ dst = out + (size_t)plane * (size_t)npix;

    const float s0 = v6ws[0], s1 = v6ws[1], s2 = v6ws[2];
    const float s3 = v6ws[3], s4 = v6ws[4], s5 = v6ws[5];

    const bool lo = (lane < 16);
    v2f bm1; bm1.x = lo ? s0 : s2;   bm1.y = lo ? s1 : s3;   // K0/K1 | K2/K3
    v2f bm2; bm2.x = lo ? s4 : 0.0f; bm2.y = lo ? s5 : 0.0f; // K4/K5 | K6/K7(pad)

    const int  laneSub = lane & 15;
    const int  perm    = (lane < 8) ? lane
                       : (lane < 16) ? lane + 8
                       : (lane < 24) ? lane - 8
                       : lane;
    const bool useT1   = (lane & 8) != 0;

    for (long base = wid0 * 32; base < (long)npix; base += stridePix) {
        const float* q0 = src + (size_t)(base + laneSub) * 6;   // tile 0 pixel row
        const float* q1 = q0 + 96;                              // tile 1 (16 px * 6)

        // Issue all loads up front (8-byte aligned b64s), one wait, then WMMAs.
        v2f a1_0 = *(const v2f*)(q0 + (lo ? 0 : 2));
        v2f a1_1 = *(const v2f*)(q1 + (lo ? 0 : 2));
        v2f a2_0; a2_0.x = 0.0f; a2_0.y = 0.0f;
        v2f a2_1; a2_1.x = 0.0f; a2_1.y = 0.0f;
        if (lo) {
            a2_0 = *(const v2f*)(q0 + 4);
            a2_1 = *(const v2f*)(q1 + 4);
        }

        // Prefetch the next grid-stride chunk into L2 (streaming hint).
        {
            long nb = base + stridePix;
            if (nb < (long)npix)
                __builtin_prefetch(src + (size_t)(nb + laneSub) * 6, 0, 0);
        }

        v8f c0 = {};
        c0 = __builtin_amdgcn_wmma_f32_16x16x4_f32(false, a1_0, false, bm1, (short)0, c0, false, false);
        c0 = __builtin_amdgcn_wmma_f32_16x16x4_f32(false, a2_0, false, bm2, (short)0, c0, false, false);
        v8f c1 = {};
        c1 = __builtin_amdgcn_wmma_f32_16x16x4_f32(false, a1_1, false, bm1, (short)0, c1, false, false);
        c1 = __builtin_amdgcn_wmma_f32_16x16x4_f32(false, a2_1, false, bm2, (short)0, c1, false, false);

        // Extract this lane's row value from each fragment (r = lane & 7).
        const int r = lane & 7;
        float vA = c0[0];
        vA = (r == 1) ? c0[1] : vA;
        vA = (r == 2) ? c0[2] : vA;
        vA = (r == 3) ? c0[3] : vA;
        vA = (r == 4) ? c0[4] : vA;
        vA = (r == 5) ? c0[5] : vA;
        vA = (r == 6) ? c0[6] : vA;
        vA = (r == 7) ? c0[7] : vA;
        float vB = c1[0];
        vB = (r == 1) ? c1[1] : vB;
        vB = (r == 2) ? c1[2] : vB;
        vB = (r == 3) ? c1[3] : vB;
        vB = (r == 4) ? c1[4] : vB;
        vB = (r == 5) ? c1[5] : vB;
        vB = (r == 6) ? c1[6] : vB;
        vB = (r == 7) ? c1[7] : vB;

        // One full-EXEC contiguous 128B store for 32 pixels.
        dst[(size_t)base + perm] = useT1 ? vB : vA;
    }
}

// ---------------------------------------------------------------------------
// Launch
// ---------------------------------------------------------------------------
extern "C" void kernel_launch(void* const* d_in, const int* in_sizes, int n_in,
                              void* d_out, int out_size, void* d_ws, size_t ws_size,
                              hipStream_t stream) {
    const float* vel   = (const float*)d_in[0];
    const float* Lsx   = (const float*)d_in[1];
    const float* Lsy   = (const float*)d_in[2];
    // d_in[3] = f12 : only its shape matters; unused.
    const float* ln0_g = (const float*)d_in[4];
    const float* ln0_b = (const float*)d_in[5];
    const float* Wq    = (const float*)d_in[6];
    const float* bq    = (const float*)d_in[7];
    const float* Wk    = (const float*)d_in[8];
    const float* bk    = (const float*)d_in[9];
    const float* Wv    = (const float*)d_in[10];
    const float* bv    = (const float*)d_in[11];
    const float* Wo    = (const float*)d_in[12];
    const float* bo    = (const float*)d_in[13];
    const float* ln1_g = (const float*)d_in[14];
    const float* ln1_b = (const float*)d_in[15];
    const float* Wi    = (const float*)d_in[16];
    const float* bi    = (const float*)d_in[17];
    const float* Wp    = (const float*)d_in[18];
    const float* bp    = (const float*)d_in[19];
    const float* ln2_g = (const float*)d_in[20];
    const float* ln2_b = (const float*)d_in[21];

    float* out = (float*)d_out;
    float* v6  = (float*)d_ws; // 6 floats of scratch

    tiny_transformer_kernel<<<1, 32, 0, stream>>>(
        vel, ln0_g, ln0_b, Wq, bq, Wk, bk, Wv, bv, Wo, bo,
        ln1_g, ln1_b, Wi, bi, Wp, bp, ln2_g, ln2_b, v6);

    const int npix = in_sizes[1] / 6;                 // 1536*1536
    const long pairsPerPlane = (long)npix / 32;       // 32 pixels per wave-iter
    const int wavesPerBlock = 8;                      // 256 threads, wave32
    long blocksL = (pairsPerPlane + wavesPerBlock - 1) / wavesPerBlock;
    int blocksX = (blocksL > 2048) ? 2048 : (int)blocksL;
    if (blocksX < 1) blocksX = 1;

    dim3 grid((unsigned)blocksX, 2, 1);               // y = plane (Lsx / Lsy)
    field_project_kernel<<<grid, 256, 0, stream>>>(Lsx, Lsy, v6, out, npix);
}